// RankingLoss_90409061581288
// MI455X (gfx1250) — compile-verified
//
#include <hip/hip_runtime.h>
#include <cstdint>

// ---------------------------------------------------------------------------
// DeepHit-style ranking loss for MI455X (gfx1250).
// HBM-bound: ~155 MB effective traffic (survival + weights + rare hazard
// gathers) -> ~6.7 us floor at 23.3 TB/s. One 256-thread block per batch.
// CDNA5 paths used: global_load_async_to_lds_b128 (+ s_wait_asynccnt) for
// LDS staging, v_wmma_f32_16x16x4_f32 for the per-block 16x16 reduction.
// ---------------------------------------------------------------------------

#define EPSF 1e-6f

typedef __attribute__((ext_vector_type(2))) float v2f;
typedef __attribute__((ext_vector_type(8))) float v8f;

// Async global->LDS copy of 16 bytes per active lane (tracked by ASYNCcnt).
// %0 = LDS byte address (VGPR), %1 = 64-bit global address (VGPR pair).
__device__ __forceinline__ void async_copy16(uint32_t lds_off, const void* src) {
  asm volatile("global_load_async_to_lds_b128 %0, %1, off"
               :: "v"(lds_off), "v"((unsigned long long)(uintptr_t)src)
               : "memory");
}
__device__ __forceinline__ void wait_async0() {
  asm volatile("s_wait_asynccnt 0" ::: "memory");
}

__global__ __launch_bounds__(256)
void rank_loss_kernel(const int*   __restrict__ events,     // [n,16]
                      const float* __restrict__ durations,  // [n,16]
                      const float* __restrict__ survival,   // [n,16,129]
                      const float* __restrict__ hazard,     // [n,16,128]
                      const float* __restrict__ weights,    // [n,16,16]
                      const float* __restrict__ cuts,       // [128]
                      float*       __restrict__ acc)        // [2]: sum, count
{
  constexpr int E = 16, K = 128, SROW = K + 1;
  // float-offsets into smem (all async-dest regions are 16B aligned)
  constexpr int OFF_SURV  = 0;     // 2064 floats  (16 x 129)
  constexpr int OFF_W     = 2064;  // 256
  constexpr int OFF_DUR   = 2320;  // 16
  constexpr int OFF_EV    = 2336;  // 16 (ints)
  constexpr int OFF_CUTS  = 2352;  // 128
  constexpr int OFF_T0    = 2480;  // 16 (ints)
  constexpr int OFF_SATT  = 2496;  // 256
  constexpr int OFF_SATTM = 2752;  // 256
  constexpr int OFF_TERM  = 3008;  // 256
  constexpr int OFF_VAL   = 3264;  // 256
  __shared__ __align__(16) float smem[3520];   // 14080 B of LDS

  const int tid = threadIdx.x;
  const int b   = blockIdx.x;

  // Low 32 bits of the flat address of an LDS object == LDS byte offset.
  const uint32_t lbase = (uint32_t)(uintptr_t)smem;

  const char* gs = (const char*)survival  + (size_t)b * (E * SROW * 4);
  const char* gw = (const char*)weights   + (size_t)b * (E * E * 4);
  const char* gd = (const char*)durations + (size_t)b * (E * 4);
  const char* ge = (const char*)events    + (size_t)b * (E * 4);
  const char* gc = (const char*)cuts;

  // --- async staging: 516 + 64 + 4 + 4 + 32 b128 chunks -------------------
  for (int c = tid; c < 516; c += 256)
    async_copy16(lbase + OFF_SURV * 4u + (uint32_t)c * 16u, gs + (size_t)c * 16);
  if (tid < 64) async_copy16(lbase + OFF_W   * 4u + (uint32_t)tid * 16u, gw + (size_t)tid * 16);
  if (tid < 4)  async_copy16(lbase + OFF_DUR * 4u + (uint32_t)tid * 16u, gd + (size_t)tid * 16);
  if (tid < 4)  async_copy16(lbase + OFF_EV  * 4u + (uint32_t)tid * 16u, ge + (size_t)tid * 16);
  if (tid < 32) async_copy16(lbase + OFF_CUTS* 4u + (uint32_t)tid * 16u, gc + (size_t)tid * 16);
  wait_async0();
  __syncthreads();

  float* s_surv  = smem + OFF_SURV;
  float* s_w     = smem + OFF_W;
  float* s_dur   = smem + OFF_DUR;
  int*   s_ev    = (int*)(smem + OFF_EV);
  float* s_cuts  = smem + OFF_CUTS;
  int*   s_t0    = (int*)(smem + OFF_T0);
  float* s_SatT  = smem + OFF_SATT;
  float* s_SatTM = smem + OFF_SATTM;
  float* s_term  = smem + OFF_TERM;
  float* s_val   = smem + OFF_VAL;

  // --- searchsorted bin per column j --------------------------------------
  if (tid < E) {
    const float d = s_dur[tid];
    int cnt = 0;
    #pragma unroll 8
    for (int k = 0; k < K; ++k) cnt += (s_cuts[k] <= d) ? 1 : 0;
    int t0 = cnt - 1;
    s_t0[tid] = (t0 < 0) ? 0 : t0;
  }
  __syncthreads();

  const int i = tid >> 4;   // row (event index of survival curve)
  const int j = tid & 15;   // column (event index of duration)

  {
    const int   t0 = s_t0[j];
    const int   t1 = (t0 + 1 > K - 1) ? (K - 1) : (t0 + 1);
    const float T0 = s_cuts[t0];
    const float T1 = s_cuts[t1];
    const float S0 = s_surv[i * SROW + t0];
    const float S1 = s_surv[i * SROW + t1];
    const float dT = T1 - T0;
    float hstar;
    if (dT > 0.f) {
      hstar = (logf(S0 + EPSF) - logf(S1 + EPSF)) / dT;
    } else {
      // rare path (t0 == K-1): gather hazard directly from HBM
      hstar = hazard[((size_t)b * E + i) * K + t0];
    }
    const float dj    = s_dur[j];
    const float t_eps = (s_cuts[K - 1] - s_cuts[0]) / s_cuts[K - 1];
    float tm = dj - t_eps; if (tm < 0.f) tm = 0.f;
    s_SatT [i * 16 + j] = S0 * expf(-(dj - T0) * hstar);
    s_SatTM[i * 16 + j] = S0 * expf(-(tm - T0) * hstar);
  }
  __syncthreads();

  {
    const float dg_i  = s_SatT [i * 16 + i];
    const float dg_j  = s_SatT [j * 16 + j];
    const float dg2_j = s_SatTM[j * 16 + j];
    const float dS1 = dg_i - s_SatT[j * 16 + i];
    const float dS2 = s_SatTM[i * 16 + j] - dg2_j;
    const float dS3 = s_SatT [i * 16 + j] - dg_j;
    const float evi = (float)s_ev[i];
    const float evj = (float)s_ev[j];
    const float comp = (s_dur[j] > s_dur[i]) ? 1.f : 0.f;  // relu(sign(dj-di))
    const float A1 = evi * comp;
    const float A2 = A1 * evj;
    const float A3 = A1 * (1.f - evj);
    s_term[tid] = s_w[i * 16 + j] *
                  (A1 * expf(dS1) + A2 * expf(dS2) + A3 * expf(dS3));
    // A1+A2+A3 == 2*A1, so "valid" <=> A1 > 0
    s_val[tid] = (A1 > 0.f) ? 1.f : 0.f;
  }
  __syncthreads();

  // --- WMMA reduction of the 16x16 term/valid matrices (wave 0 only) ------
  // D = ones(16x4) x B(4x16) + C, chained over 4 row-chunks: every element of
  // D equals a column sum, so sum(D)/16 == sum(matrix). This is exact for any
  // bijective element->slot placement in B because A is all-ones.
  if (tid < 32) {
    v2f a; a.x = 1.f; a.y = 1.f;
    v8f accT = {0.f, 0.f, 0.f, 0.f, 0.f, 0.f, 0.f, 0.f};
    v8f accV = accT;
    const int col  = tid & 15;
    const int half = tid >> 4;
    #pragma unroll
    for (int c = 0; c < 4; ++c) {
      const int r0 = c * 4 + half * 2;
      v2f bt, bv;
      bt.x = s_term[(r0 + 0) * 16 + col];
      bt.y = s_term[(r0 + 1) * 16 + col];
      bv.x = s_val [(r0 + 0) * 16 + col];
      bv.y = s_val [(r0 + 1) * 16 + col];
      accT = __builtin_amdgcn_wmma_f32_16x16x4_f32(false, a, false, bt,
                                                   (short)0, accT, false, false);
      accV = __builtin_amdgcn_wmma_f32_16x16x4_f32(false, a, false, bv,
                                                   (short)0, accV, false, false);
    }
    float st = 0.f, sv = 0.f;
    #pragma unroll
    for (int k = 0; k < 8; ++k) { st += accT[k]; sv += accV[k]; }
    #pragma unroll
    for (int off = 16; off > 0; off >>= 1) {
      st += __shfl_down(st, off, 32);
      sv += __shfl_down(sv, off, 32);
    }
    if (tid == 0) {
      atomicAdd(&acc[0], st * 0.0625f);   // /16: D rows replicate col sums
      atomicAdd(&acc[1], sv * 0.0625f);   // exact: sv is 16 * integer count
    }
  }
}

__global__ void init_ws_kernel(float* acc) {
  if (threadIdx.x < 2) acc[threadIdx.x] = 0.f;
}

__global__ void finalize_kernel(const float* __restrict__ acc,
                                float* __restrict__ out) {
  const float s = acc[0], v = acc[1];
  out[0] = (v > 0.f) ? s / fmaxf(v, 1.f) : 0.f;
}

extern "C" void kernel_launch(void* const* d_in, const int* in_sizes, int n_in,
                              void* d_out, int out_size, void* d_ws, size_t ws_size,
                              hipStream_t stream) {
  const int*   events    = (const int*)  d_in[0];
  const float* durations = (const float*)d_in[1];
  const float* survival  = (const float*)d_in[2];
  const float* hazard    = (const float*)d_in[3];
  const float* weights   = (const float*)d_in[4];
  const float* cuts      = (const float*)d_in[5];
  const int n = in_sizes[0] / 16;         // events is [n,16]
  float* acc = (float*)d_ws;              // [0]=sum(loss), [1]=num_valid

  hipLaunchKernelGGL(init_ws_kernel, dim3(1), dim3(32), 0, stream, acc);
  hipLaunchKernelGGL(rank_loss_kernel, dim3(n), dim3(256), 0, stream,
                     events, durations, survival, hazard, weights, cuts, acc);
  hipLaunchKernelGGL(finalize_kernel, dim3(1), dim3(1), 0, stream,
                     acc, (float*)d_out);
}